// HyperScatteringModule_3564822856145
// MI455X (gfx1250) — compile-verified
//
#include <hip/hip_runtime.h>
#include <hip/hip_bf16.h>

// -----------------------------------------------------------------------------
// HyperScattering on MI455X (gfx1250, wave32).
//
// Phase 1 (dominant): 16 diffusion levels of gather + scatter-add over 1.6M
// incidences x 128ch. Working set (~65MB/level) lives in the 192MB L2, so we
// use one wave per incidence: float4 gather per lane + 4 f32 global atomics.
// Phase 2: wavelet combine = (6x17) @ (17 x N*C) GEMM done with
// V_WMMA_F32_16X16X4_F32 (K padded 17->20, 5 WMMA per 16-col tile), fused SiLU
// and interleaved store into the n,(w*128+f) output layout.
//
// Workspace (floats): Dv[N] | Dhe[E] | nodesL[17*N*128] | edgesL[17*E*128]
//   = ~653 MB for N=50000, E=25000.
// -----------------------------------------------------------------------------

typedef __attribute__((ext_vector_type(2))) float v2f;
typedef __attribute__((ext_vector_type(8))) float v8f;

#define C128 128

__device__ __forceinline__ void atomAddF(float* p, float v) {
  unsafeAtomicAdd(p, v);   // hardware global_atomic_add_f32 path
}

// ---- degree accumulation: Dv[src] += w[edg]; Dhe[edg] += 1 ------------------
__global__ void __launch_bounds__(256)
deg_kernel(const int* __restrict__ src, const int* __restrict__ edg,
           const float* __restrict__ hw, float* __restrict__ Dv,
           float* __restrict__ Dhe, int nnz) {
  int i = blockIdx.x * blockDim.x + threadIdx.x;
  int stride = gridDim.x * blockDim.x;
  for (; i < nnz; i += stride) {
    int s = src[i];
    int e = edg[i];
    atomAddF(&Dv[s], hw[e]);
    atomAddF(&Dhe[e], 1.0f);
  }
}

// ---- x -> (x>0 ? 1/x : 0), in place ----------------------------------------
__global__ void __launch_bounds__(256)
invert_kernel(float* __restrict__ d, int n) {
  int i = blockIdx.x * blockDim.x + threadIdx.x;
  if (i < n) {
    float v = d[i];
    d[i] = (v > 0.0f) ? (1.0f / v) : 0.0f;
  }
}

// ---- one half-step of diffusion: dst[sidx] += ginv[g]*sinv[s]*srcF[gidx] ----
// One wave32 per incidence; lane covers 4 channels (float4 load + 4 atomics).
__global__ void __launch_bounds__(256)
scatter_kernel(const float* __restrict__ gatherF,  // [Sg, 128]
               const float* __restrict__ ginv,     // inv-degree of gather side
               const int* __restrict__ gidx,
               float* __restrict__ scatF,          // [Sd, 128] (pre-zeroed)
               const float* __restrict__ sinv,     // inv-degree of scatter side
               const int* __restrict__ sidx,
               int nnz) {
  const int lane = threadIdx.x & 31;                           // wave32
  long long w = (long long)blockIdx.x * (blockDim.x >> 5) + (threadIdx.x >> 5);
  const long long nw = (long long)gridDim.x * (blockDim.x >> 5);
  for (; w < (long long)nnz; w += nw) {
    // speculative prefetch of next index pair -> global_prefetch_b8
    __builtin_prefetch(gidx + w + nw, 0, 0);
    __builtin_prefetch(sidx + w + nw, 0, 0);
    const int g = gidx[w];
    const int s = sidx[w];
    const float sc = ginv[g] * sinv[s];
    const float4* gp = (const float4*)(gatherF + (size_t)g * C128);
    float4 v = gp[lane];
    float* sp = scatF + (size_t)s * C128 + lane * 4;
    atomAddF(sp + 0, v.x * sc);
    atomAddF(sp + 1, v.y * sc);
    atomAddF(sp + 2, v.z * sc);
    atomAddF(sp + 3, v.w * sc);
  }
}

// ---- wavelet combine + SiLU via V_WMMA_F32_16X16X4_F32 ----------------------
// L: [17, NC] stacked levels. W6: [6,17]. out: [NC/128, 768] with
// out[node, r*128 + f] = silu( sum_j W6[r][j] * L[j, node*128+f] ).
// K padded 17 -> 20 (5 chunks of 4); M padded 6 -> 16 with zero rows.
__global__ void __launch_bounds__(256)
wavelet_silu_wmma(const float* __restrict__ L,
                  const float* __restrict__ W6,
                  float* __restrict__ out,
                  long long NC) {
  const int lane = threadIdx.x & 31;
  const int half = lane >> 4;     // 0: K pair {0,1}, 1: K pair {2,3}
  const int m = lane & 15;        // A row / B,D column within tile

  // A fragments (invariant across tiles): 16x4 f32 per chunk = v2f per lane.
  v2f a[5];
#pragma unroll
  for (int kk = 0; kk < 5; ++kk) {
    const int kb = kk * 4 + half * 2;
    float ax = 0.0f, ay = 0.0f;
    if (m < 6) {
      if (kb < 17)     ax = W6[m * 17 + kb];
      if (kb + 1 < 17) ay = W6[m * 17 + kb + 1];
    }
    a[kk].x = ax;
    a[kk].y = ay;
  }

  const long long tiles = NC >> 4;  // 16 columns per tile
  long long t = (long long)blockIdx.x * (blockDim.x >> 5) + (threadIdx.x >> 5);
  const long long nw = (long long)gridDim.x * (blockDim.x >> 5);

  for (; t < tiles; t += nw) {
    const long long col = (t << 4) + m;   // this lane's column for B and D
    v8f c = {};
#pragma unroll
    for (int kk = 0; kk < 5; ++kk) {
      const int kb = kk * 4 + half * 2;
      // clamp row index in-bounds, zero the padded K rows via multiplier
      const int k0 = (kb < 17) ? kb : 16;
      const int k1 = (kb + 1 < 17) ? (kb + 1) : 16;
      const float m0 = (kb < 17) ? 1.0f : 0.0f;
      const float m1 = (kb + 1 < 17) ? 1.0f : 0.0f;
      v2f b;
      b.x = L[(long long)k0 * NC + col] * m0;
      b.y = L[(long long)k1 * NC + col] * m1;
      // D = A*B + C ; 8 args: (neg_a, A, neg_b, B, c_mod, C, reuse_a, reuse_b)
      c = __builtin_amdgcn_wmma_f32_16x16x4_f32(false, a[kk], false, b,
                                                (short)0, c, false, false);
    }
    // D layout: lanes 0..15 hold rows 0..7 in c[0..7]; we need rows 0..5.
    if (half == 0) {
      const long long node = col >> 7;          // col / 128
      const int f = (int)(col & 127);
      float* o = out + node * 768 + f;
#pragma unroll
      for (int r = 0; r < 6; ++r) {
        const float v = c[r];
        o[r * 128] = v / (1.0f + expf(-v));     // silu
      }
    }
  }
}

// -----------------------------------------------------------------------------
extern "C" void kernel_launch(void* const* d_in, const int* in_sizes, int n_in,
                              void* d_out, int out_size, void* d_ws, size_t ws_size,
                              hipStream_t stream) {
  const float* x     = (const float*)d_in[0];  // [N, 128]
  const float* hw    = (const float*)d_in[1];  // [E]
  const float* hattr = (const float*)d_in[2];  // [E, 128]
  const float* W6    = (const float*)d_in[3];  // [6, 17]
  const int*   hidx  = (const int*)d_in[4];    // [2, NNZ]

  const size_t N   = (size_t)in_sizes[0] / C128;
  const size_t E   = (size_t)in_sizes[1];
  const size_t NNZ = (size_t)in_sizes[4] / 2;
  const int LEVELS = 16;

  const int* src = hidx;        // row 0
  const int* edg = hidx + NNZ;  // row 1

  // ---- workspace layout ----
  float* ws  = (float*)d_ws;
  float* Dv  = ws;              // [N]  -> Dv_inv in place
  float* Dhe = ws + N;          // [E]  -> Dhe_inv in place
  size_t degPad = ((N + E + 255) / 256) * 256;
  float* nodesL = ws + degPad;                       // [17, N, 128]
  float* edgesL = nodesL + (size_t)17 * N * C128;    // [17, E, 128]

  const size_t nodeBytes = N * C128 * sizeof(float);
  const size_t edgeBytes = E * C128 * sizeof(float);

  // ---- degrees ----
  hipMemsetAsync(Dv, 0, (N + E) * sizeof(float), stream);
  deg_kernel<<<2048, 256, 0, stream>>>(src, edg, hw, Dv, Dhe, (int)NNZ);
  {
    int tot = (int)(N + E);
    invert_kernel<<<(tot + 255) / 256, 256, 0, stream>>>(Dv, tot); // Dv,Dhe contiguous
  }

  // ---- level 0 = inputs ----
  hipMemcpyAsync(nodesL, x, nodeBytes, hipMemcpyDeviceToDevice, stream);
  hipMemcpyAsync(edgesL, hattr, edgeBytes, hipMemcpyDeviceToDevice, stream);

  // ---- 16 diffusion levels ----
  const int wavesPerBlock = 256 / 32;  // 8
  int scatBlocks = (int)((NNZ + wavesPerBlock - 1) / wavesPerBlock);
  for (int lev = 1; lev <= LEVELS; ++lev) {
    float* prevN = nodesL + (size_t)(lev - 1) * N * C128;
    float* curE  = edgesL + (size_t)lev * E * C128;
    float* curN  = nodesL + (size_t)lev * N * C128;

    // node -> edge: curE[e] += Dhe_inv[e] * Dv_inv[v] * prevN[v]
    hipMemsetAsync(curE, 0, edgeBytes, stream);
    scatter_kernel<<<scatBlocks, 256, 0, stream>>>(prevN, Dv, src,
                                                   curE, Dhe, edg, (int)NNZ);
    // edge -> node: curN[v] += Dv_inv[v] * Dhe_inv[e] * curE[e]
    hipMemsetAsync(curN, 0, nodeBytes, stream);
    scatter_kernel<<<scatBlocks, 256, 0, stream>>>(curE, Dhe, edg,
                                                   curN, Dv, src, (int)NNZ);
  }

  // ---- wavelet combine + silu (WMMA), straight into d_out ----
  float* outNodes = (float*)d_out;                    // [N, 768]
  float* outEdges = outNodes + N * 6 * C128;          // [E, 768]

  const long long NCn = (long long)N * C128;
  const long long NCe = (long long)E * C128;
  const long long tilesN = NCn >> 4;
  const long long tilesE = NCe >> 4;
  int blkN = (int)((tilesN + wavesPerBlock - 1) / wavesPerBlock);
  int blkE = (int)((tilesE + wavesPerBlock - 1) / wavesPerBlock);

  wavelet_silu_wmma<<<blkN, 256, 0, stream>>>(nodesL, W6, outNodes, NCn);
  wavelet_silu_wmma<<<blkE, 256, 0, stream>>>(edgesL, W6, outEdges, NCe);
}